// VectorQuantizer_56324201120199
// MI455X (gfx1250) — compile-verified
//
#include <hip/hip_runtime.h>

// VQ-VAE vector quantizer for MI455X (gfx1250, wave32).
//   x:        (16, 256, 32, 32) f32  -> X  = (16384, 256) after (b,h,w,c) flatten
//   codebook: (8192, 256) f32
// dist matmul 16384x8192x256 = 68.7 GFLOP, compute-bound (~50-100x over the
// 23.3 TB/s HBM roofline; codebook re-streams from the 192MB L2). Exact-f32
// path via V_WMMA_F32_16X16X4_F32 because codebook magnitudes (~1e-4) make
// the argmin bf16/f16-fragile. Codebook tiles staged into LDS by the Tensor
// Data Mover (tensor_load_to_lds + s_wait_tensorcnt) with hardware row
// padding (256 DWORDs interval, 4 DWORDs pad -> 260-float rows, bank-clean).

#define NUM_EMB 8192
#define DIM 256
#define NROWS 16384                       // 16*32*32
#define TOTAL_ELEMS (16 * 256 * 32 * 32)  // 4194304
#define BETA 0.25f
#define LDSROW 260   // 256 + 4 pad floats: 16B-aligned rows, conflict-free banks

typedef float v2f __attribute__((ext_vector_type(2)));
typedef float v8f __attribute__((ext_vector_type(8)));
typedef int   i32x4 __attribute__((ext_vector_type(4)));
typedef int   i32x8 __attribute__((ext_vector_type(8)));

#if defined(__has_builtin)
#if __has_builtin(__builtin_amdgcn_tensor_load_to_lds_d2) || \
    __has_builtin(__builtin_amdgcn_tensor_load_to_lds)
#define VQ_HAVE_TDM 1
#endif
#endif

// ---------------------------------------------------------------------------
// Kernel 0: cbnorm[k] = sum_c codebook[k][c]^2   (one wave32 per code)
// ---------------------------------------------------------------------------
__global__ __launch_bounds__(256) void vq_cbnorm_kernel(
    const float* __restrict__ cb, float* __restrict__ cbn) {
  const int g    = (int)(blockIdx.x * blockDim.x + threadIdx.x);
  const int wv   = g >> 5;
  const int lane = threadIdx.x & 31;
  if (wv >= NUM_EMB) return;
  const float* row = cb + (size_t)wv * DIM;
  float s = 0.f;
#pragma unroll
  for (int c = 0; c < DIM / 32; ++c) {
    float v = row[lane + 32 * c];
    s = fmaf(v, v, s);
  }
#pragma unroll
  for (int off = 16; off; off >>= 1) s += __shfl_xor(s, off, 32);
  if (lane == 0) cbn[wv] = s;
}

#ifdef VQ_HAVE_TDM
// Issue a TDM load of one codebook chunk (16 rows x 256 f32, row-major) into
// LDS at byte offset lds_off, with 16B padding appended after every 1024B row
// (-> LDSROW=260 float row pitch). Wave-level op; EXEC ignored; TENSORcnt.
__device__ __forceinline__ void tdm_load_cb_chunk(const float* __restrict__ cb,
                                                  int ch, unsigned lds_off) {
  const unsigned long long ga =
      (unsigned long long)(uintptr_t)(cb + (size_t)ch * 16 * DIM);
  // D# group 0: count=1 | lds_addr | global_addr[56:0] | type=2
  i32x4 g0 = {1, (int)lds_off, (int)(unsigned)ga,
              (int)((((unsigned)(ga >> 32)) & 0x1FFFFFFu) | 0x80000000u)};
  // D# group 1:
  //  w0: data_size=4B (2<<16) | pad_enable (1<<20) | pad_interval=256dw (7<<22)
  //      | pad_amount=4dw (3<<25); workgroup_mask=0
  //  w1[31:16]=tensor_dim0.lo16=256   w2[15:0]=tensor_dim0.hi16=0
  //  w2[31:16]=tensor_dim1.lo16=8192  w3[15:0]=tensor_dim1.hi16=0
  //  w3[31:16]=tile_dim0=256          w4[15:0]=tile_dim1=16, tile_dim2=0
  //  w5..w6[15:0]=tensor_dim0_stride=256; dim1_stride unused (2D)
  i32x8 g1 = {(int)((2u << 16) | (1u << 20) | (7u << 22) | (3u << 25)),
              (int)(256u << 16),
              (int)(8192u << 16),
              (int)(256u << 16),
              16,
              256,
              0,
              0};
#if __has_builtin(__builtin_amdgcn_tensor_load_to_lds_d2)
  __builtin_amdgcn_tensor_load_to_lds_d2(g0, g1, 0);
#else
  i32x4 z4 = {0, 0, 0, 0};
#if __clang_major__ >= 23
  i32x8 z8 = {0, 0, 0, 0, 0, 0, 0, 0};
  __builtin_amdgcn_tensor_load_to_lds(g0, g1, z4, z4, z8, 0);
#else
  __builtin_amdgcn_tensor_load_to_lds(g0, g1, z4, z4, 0);
#endif
#endif
}
#endif  // VQ_HAVE_TDM

// ---------------------------------------------------------------------------
// Kernel 1: fused S = X*C^T via WMMA f32 16x16x4 + per-row argmin of
//           (||c_k||^2 - 2*S[n,k])   (||x||^2 is argmin-invariant, dropped)
// 4 waves / 128 threads per WG, 16 x-rows per wave (A in registers),
// codebook streamed in 16-code chunks through double-buffered LDS via TDM.
// Two independent accumulators break the WMMA RAW chain (2-way XDL ILP).
// ---------------------------------------------------------------------------
__global__ __launch_bounds__(128) void vq_argmin_kernel(
    const float* __restrict__ x, const float* __restrict__ cb,
    const float* __restrict__ cbn, int* __restrict__ ids) {
  __shared__ float CBs[2][16 * LDSROW];

  const int tid  = threadIdx.x;
  const int lane = tid & 31;
  const int wave = tid >> 5;
  const int half = lane >> 4;  // 0: K lanes {0,1}, 1: K lanes {2,3}
  const int l15  = lane & 15;
  const int klo  = half << 1;

  const int rowbase = blockIdx.x * 64 + wave * 16;
  const int m       = rowbase + l15;  // global x-row this lane feeds A with
  // x_[n, c] lives at x[(n>>10)*262144 + (n&1023) + c*1024]
  const float* xb = x + (size_t)(m >> 10) * (DIM * 1024) + (m & 1023);

  // A-fragments (16x4 f32 layout): lane holds X[m, 4i+klo], X[m, 4i+klo+1]
  v2f a[DIM / 4];
#pragma unroll
  for (int i = 0; i < DIM / 4; ++i) {
    const int c = 4 * i + klo;
    a[i].x = xb[c * 1024];
    a[i].y = xb[c * 1024 + 1024];
  }

  float best[8];
  int   bidx[8];
#pragma unroll
  for (int v = 0; v < 8; ++v) { best[v] = 3.4e38f; bidx[v] = 0; }

#ifdef VQ_HAVE_TDM
  const unsigned lds0   = (unsigned)(uintptr_t)(void*)&CBs[0][0];
  const unsigned lds1   = (unsigned)(uintptr_t)(void*)&CBs[1][0];
  const bool     issuer = (wave == 0);  // wave-uniform: one TDM op per chunk/WG
  if (issuer) {
    tdm_load_cb_chunk(cb, 0, lds0);
    __builtin_amdgcn_s_wait_tensorcnt(0);
  }
#else
  // Fallback loader: 16 codes x 256 floats -> LDS, 128 threads, float4 moves
  const int lr = tid >> 3;        // row 0..15
  const int lc = (tid & 7) * 32;  // 32 floats per thread per row
  auto load_chunk = [&](int ch, int buf) {
    const float* src = cb + (size_t)(ch * 16 + lr) * DIM + lc;
    float*       dst = &CBs[buf][lr * LDSROW + lc];
#pragma unroll
    for (int j = 0; j < 8; ++j) {
      float4 t = *(const float4*)(src + 4 * j);
      *(float4*)(dst + 4 * j) = t;
    }
  };
  load_chunk(0, 0);
#endif
  __syncthreads();

  for (int ch = 0; ch < NUM_EMB / 16; ++ch) {
    const int cur = ch & 1;
#ifdef VQ_HAVE_TDM
    if (issuer && ch + 1 < NUM_EMB / 16)
      tdm_load_cb_chunk(cb, ch + 1, cur ? lds0 : lds1);
#else
    if (ch + 1 < NUM_EMB / 16) load_chunk(ch + 1, cur ^ 1);
#endif

    // B-fragments (4x16 f32): lane holds C[n=l15][kk+klo], C[n][kk+klo+1]
    const float* brow = &CBs[cur][l15 * LDSROW + klo];
    v8f acc0 = {0.f, 0.f, 0.f, 0.f, 0.f, 0.f, 0.f, 0.f};
    v8f acc1 = {0.f, 0.f, 0.f, 0.f, 0.f, 0.f, 0.f, 0.f};
#pragma unroll
    for (int i = 0; i < DIM / 4; i += 2) {
      v2f b0 = *(const v2f*)(brow + 4 * i);
      v2f b1 = *(const v2f*)(brow + 4 * i + 4);
      acc0 = __builtin_amdgcn_wmma_f32_16x16x4_f32(false, a[i], false, b0,
                                                   (short)0, acc0, false, false);
      acc1 = __builtin_amdgcn_wmma_f32_16x16x4_f32(false, a[i + 1], false, b1,
                                                   (short)0, acc1, false, false);
    }

    const int   code = ch * 16 + l15;  // D: lane holds column N=l15, rows v+8*half
    const float cn   = cbn[code];      // L0/L2-cached, 16 distinct addrs/chunk
#pragma unroll
    for (int v = 0; v < 8; ++v) {
      float d = cn - 2.0f * (acc0[v] + acc1[v]);
      if (d < best[v]) { best[v] = d; bidx[v] = code; }  // strict < => first idx wins
    }

#ifdef VQ_HAVE_TDM
    if (issuer) __builtin_amdgcn_s_wait_tensorcnt(0);  // publish next buffer
#endif
    __syncthreads();
  }

  // reduce (val,idx) across the 16 lanes holding each row; tie -> smaller index
#pragma unroll
  for (int v = 0; v < 8; ++v) {
    float val = best[v];
    int   idx = bidx[v];
#pragma unroll
    for (int off = 1; off < 16; off <<= 1) {
      float ov = __shfl_xor(val, off, 32);
      int   oi = __shfl_xor(idx, off, 32);
      if (ov < val || (ov == val && oi < idx)) { val = ov; idx = oi; }
    }
    if (l15 == 0) ids[rowbase + half * 8 + v] = idx;
  }
}

// ---------------------------------------------------------------------------
// Kernel 2: emb_out = x + (emb - x) (straight-through forward value) and
//           per-block partial sums of (x - emb)^2. Coalesced x/emb_out HBM
//           traffic; codebook gathers hit L2 (codebook is 8 MB, L2 is 192 MB).
// ---------------------------------------------------------------------------
__global__ __launch_bounds__(256) void vq_emb_loss_kernel(
    const float* __restrict__ x, const float* __restrict__ cb,
    const int* __restrict__ ids, float* __restrict__ emb_out,
    float* __restrict__ partial) {
  const int tid = threadIdx.x;
  float s = 0.f;
#pragma unroll
  for (int j = 0; j < 4; ++j) {
    const int e    = (int)blockIdx.x * 1024 + j * 256 + tid;
    const int bimg = e >> 18;
    const int c    = (e >> 10) & 255;
    const int hw   = e & 1023;
    const int n    = (bimg << 10) | hw;
    const float xv = x[e];
    const float ev = cb[(size_t)ids[n] * DIM + c];
    emb_out[e]     = xv + (ev - xv);  // matches reference fp order
    const float d  = xv - ev;
    s = fmaf(d, d, s);
  }
  __shared__ float red[8];
  const int lane = tid & 31, wv = tid >> 5;
#pragma unroll
  for (int off = 16; off; off >>= 1) s += __shfl_xor(s, off, 32);
  if (lane == 0) red[wv] = s;
  __syncthreads();
  if (tid == 0) {
    float t = 0.f;
#pragma unroll
    for (int i = 0; i < 8; ++i) t += red[i];
    partial[blockIdx.x] = t;
  }
}

// ---------------------------------------------------------------------------
// Kernel 3: deterministic final reduction ->
//           loss = (1 + beta) * mean((x - emb)^2)
// ---------------------------------------------------------------------------
__global__ __launch_bounds__(256) void vq_loss_final_kernel(
    const float* __restrict__ partial, float* __restrict__ loss) {
  float s = 0.f;
  for (int i = threadIdx.x; i < TOTAL_ELEMS / 1024; i += 256) s += partial[i];
  __shared__ float red[8];
  const int lane = threadIdx.x & 31, wv = threadIdx.x >> 5;
#pragma unroll
  for (int off = 16; off; off >>= 1) s += __shfl_xor(s, off, 32);
  if (lane == 0) red[wv] = s;
  __syncthreads();
  if (threadIdx.x == 0) {
    float t = 0.f;
#pragma unroll
    for (int i = 0; i < 8; ++i) t += red[i];
    *loss = (1.0f + BETA) * t * (1.0f / (float)TOTAL_ELEMS);
  }
}

// ---------------------------------------------------------------------------
extern "C" void kernel_launch(void* const* d_in, const int* in_sizes, int n_in,
                              void* d_out, int out_size, void* d_ws, size_t ws_size,
                              hipStream_t stream) {
  const float* x  = (const float*)d_in[0];
  const float* cb = (const float*)d_in[1];

  // d_out: [ids (16384 int32)] [emb_out (4194304 f32)] [loss (1 f32)]
  float* outf    = (float*)d_out;
  int*   ids     = (int*)d_out;
  float* emb_out = outf + NROWS;
  float* loss    = outf + NROWS + TOTAL_ELEMS;

  // workspace: [cbnorm (8192 f32)] [partials (4096 f32)]
  float* cbn     = (float*)d_ws;
  float* partial = cbn + NUM_EMB;

  vq_cbnorm_kernel<<<NUM_EMB * 32 / 256, 256, 0, stream>>>(cb, cbn);
  vq_argmin_kernel<<<NROWS / 64, 128, 0, stream>>>(x, cb, cbn, ids);
  vq_emb_loss_kernel<<<TOTAL_ELEMS / 1024, 256, 0, stream>>>(x, cb, ids, emb_out, partial);
  vq_loss_final_kernel<<<1, 256, 0, stream>>>(partial, loss);
}